// Exphydro_84731114815948
// MI455X (gfx1250) — compile-verified
//
#include <hip/hip_runtime.h>
#include <cstdint>
#include <cstddef>

#define T_STEPS 1095
#define G_CELLS 15000
#define LENF    15
#define BLOCK   128
#define CH      3
#define ROW     (G_CELLS * CH)     /* 45000 floats per timestep row   */
#define TILE_W  (BLOCK * CH)       /* 384 floats per tile row         */
#define CHUNKS  (T_STEPS / LENF)   /* 73 chunks of exactly 15 steps   */
#define NZ      1e-5f
#define LOG2E   1.44269504088896340736f

typedef unsigned int u32x4 __attribute__((ext_vector_type(4)));
typedef int          i32x4 __attribute__((ext_vector_type(4)));
typedef int          i32x8 __attribute__((ext_vector_type(8)));
typedef float        f32x2 __attribute__((ext_vector_type(2)));

// Hardware transcendentals (TRANS pipe, co-executes with VALU)
__device__ __forceinline__ float hw_log2(float x) { return __builtin_amdgcn_logf(x); }
__device__ __forceinline__ float hw_exp2(float x) { return __builtin_amdgcn_exp2f(x); }

// ---------------------------------------------------------------------------
// Packed fp32 FMA: d = a*b + c per 32-bit lane of a 64-bit VGPR pair.
// V_PK_FMA_F32 (CDNA5 VOP3P opcode 31).
// ---------------------------------------------------------------------------
__device__ __forceinline__ f32x2 pk_fma(f32x2 a, f32x2 b, f32x2 c) {
  f32x2 d;
  asm("v_pk_fma_f32 %0, %1, %2, %3" : "=v"(d) : "v"(a), "v"(b), "v"(c));
  return d;
}

// ---------------------------------------------------------------------------
// TDM: issue one 2D tile load (LENF rows x tile_w floats) into LDS.
// D# group layout per CDNA5 ISA ch.8 (CDNA5 ISA).
// ---------------------------------------------------------------------------
__device__ __forceinline__ void tdm_load_chunk(const float* xp, int k, int g0c,
                                               unsigned lds_off, int tile_w) {
  uint64_t ga = (uint64_t)(uintptr_t)xp +
                ((uint64_t)k * LENF * ROW + (uint64_t)g0c) * 4ull;
  u32x4 grp0;
  grp0[0] = 1u;                                   // count=1, user descriptor
  grp0[1] = lds_off;                              // lds_addr (bytes)
  grp0[2] = (unsigned)(ga & 0xFFFFFFFFu);         // global_addr[31:0]
  grp0[3] = (unsigned)((ga >> 32) & 0x01FFFFFFu)  // global_addr[56:32]
          | (2u << 30);                           // type = 2 ("image")
  i32x8 grp1;
  grp1[0] = 0x00020000;                           // data_size = 4B, no iterate/pad
  grp1[1] = (int)(((unsigned)ROW & 0xFFFFu) << 16);            // tensor_dim0 lo16
  grp1[2] = (int)((((unsigned)ROW >> 16) & 0xFFFFu)            // tensor_dim0 hi16
          |        (((unsigned)T_STEPS & 0xFFFFu) << 16));     // tensor_dim1 lo16
  grp1[3] = (int)((((unsigned)T_STEPS >> 16) & 0xFFFFu)        // tensor_dim1 hi16
          |        ((unsigned)tile_w << 16));                  // tile_dim0
  grp1[4] = LENF;                                 // tile_dim1 = 15, tile_dim2 = 0
  grp1[5] = ROW;                                  // tensor_dim0_stride[31:0]
  grp1[6] = 0;
  grp1[7] = 0;
  i32x4 z4 = (i32x4)0;
#if __has_include(<hip/amd_detail/amd_gfx1250_TDM.h>)
  i32x8 z8 = (i32x8)0;                            // therock toolchain: 6-arg builtin
  __builtin_amdgcn_tensor_load_to_lds(grp0, grp1, z4, z4, z8, 0);
#else                                             // ROCm 7.2: 5-arg builtin
  __builtin_amdgcn_tensor_load_to_lds(grp0, grp1, z4, z4, 0);
#endif
}

// ---------------------------------------------------------------------------
// Normalized 15-tap gamma unit hydrograph.  The reference's
// -gammaln(a) - a*log(theta) terms are constant across taps and cancel under
// w/sum(w); we drop them and subtract max(logw) (stable, no lgammaf needed).
// ln(l + 0.5) baked in as compile-time constants.
// ---------------------------------------------------------------------------
__device__ __forceinline__ void uh_gamma(float a, float th, float* w) {
  const float logt[LENF] = {
      -0.69314718f, 0.40546511f, 0.91629073f, 1.25276297f, 1.50407740f,
       1.70474809f, 1.87180218f, 2.01490302f, 2.14006616f, 2.25129180f,
       2.35137526f, 2.44234704f, 2.52572864f, 2.60268969f, 2.67414865f };
  const float am1 = a - 1.f;
  const float ith = 1.f / th;
  float lw[LENF];
  float m = -3.0e38f;
#pragma unroll
  for (int l = 0; l < LENF; ++l) {
    lw[l] = am1 * logt[l] - ((float)l + 0.5f) * ith;
    m = fmaxf(m, lw[l]);
  }
  float s = 0.f;
#pragma unroll
  for (int l = 0; l < LENF; ++l) {
    w[l] = hw_exp2((lw[l] - m) * LOG2E);          // exp(x) = exp2(x*log2e)
    s += w[l];
  }
  const float inv = 1.f / s;
#pragma unroll
  for (int l = 0; l < LENF; ++l) w[l] *= inv;
}

__global__ __launch_bounds__(BLOCK) void exphydro_kernel(
    const float* __restrict__ x_phy, const float* __restrict__ params,
    float* __restrict__ out) {
  __shared__ float tile[2][LENF * TILE_W];        // 2 x 23040 B = 45 KB

  const int  tid   = threadIdx.x;
  const int  g0    = blockIdx.x * BLOCK;
  const int  g     = g0 + tid;
  const bool valid = (g < G_CELLS);
  const int  gp    = valid ? g : (G_CELLS - 1);
  const bool wave0 = (tid >> 5) == 0;
  const int  tile_w = min(TILE_W, ROW - g0 * CH);

  // ---- per-cell parameters (4 x b128) ----
  float p[16];
  const float* pr = params + (size_t)gp * 16;
#pragma unroll
  for (int i = 0; i < 16; ++i) p[i] = pr[i];
#define DS(i, lo, hi) (p[i] * ((hi) - (lo)) + (lo))
  const float ddf   = DS(0, 0.f, 40.f);
  const float Tbm   = DS(1, -2.f, 3.f);
  const float wrf   = DS(2, 0.f, 0.5f);
  const float Tbf   = DS(3, -5.f, 2.f);
  const float Kf    = DS(4, 0.f, 5.f);
  const float exp_fe= DS(5, 0.f, 1.f);
  const float ETe   = DS(6, 0.f, 1.f);
  const float c_run = DS(7, 0.f, 1.f);
  const float c_v2p = DS(8, 1e-5f, 0.02f);
  const float c_vad = DS(9, 0.f, 0.1f);
  const float c_phr = DS(10, 1e-5f, 0.01f);
  const float vmax  = DS(11, 0.001f, 500.f);
  const float a1    = DS(12, 0.3f, 20.f);
  const float b1    = DS(13, 0.01f, 5.f);
  const float a2    = DS(14, 0.5f, 13.f);
  const float b2    = DS(15, 0.15f, 1.5f);
#undef DS
  const float inv_vmax = 1.f / vmax;

  // ---- routing filters, packed (w_surf, w_delay) per tap ----
  float w1[LENF], w2[LENF];
  uh_gamma(a1, b1, w1);
  uh_gamma(a2, b2, w2);
  f32x2 wp[LENF];
#pragma unroll
  for (int l = 0; l < LENF; ++l) { wp[l].x = w1[l]; wp[l].y = w2[l]; }

  // ---- state + packed conv scatter-ring (registers, const-indexed) ----
  float sog = NZ, wis = NZ, vad = NZ, phr = NZ;
  f32x2 acc[LENF];
#pragma unroll
  for (int i = 0; i < LENF; ++i) acc[i] = (f32x2)0.f;

  // ---- TDM double-buffered pipeline over 73 chunks of 15 timesteps ----
  if (wave0)
    tdm_load_chunk(x_phy, 0, g0 * CH, (unsigned)(uintptr_t)&tile[0][0], tile_w);

  for (int k = 0; k < CHUNKS; ++k) {
    if (wave0) __builtin_amdgcn_s_wait_tensorcnt(0);   // chunk k resident
    __syncthreads();                                   // visible to all waves
    if (wave0 && (k + 1 < CHUNKS))                     // prefetch chunk k+1
      tdm_load_chunk(x_phy, k + 1, g0 * CH,
                     (unsigned)(uintptr_t)&tile[(k + 1) & 1][0], tile_w);

    const float* buf = &tile[k & 1][0];
    const int tbase = k * LENF;
#pragma unroll
    for (int j = 0; j < LENF; ++j) {
      const float pcp = buf[j * TILE_W + tid * CH + 0];
      const float tm  = buf[j * TILE_W + tid * CH + 1];
      const float pet = buf[j * TILE_W + tid * CH + 2];

      // snow / freeze / melt bucket
      const float rain = (tm >= 0.f) ? pcp : 0.f;
      const float snow = pcp - rain;
      // pow(x,e) with x>=1e-5, e in [0,1]: hardware exp2/log2 on TRANS pipe
      const float pot_freeze =
          Kf * hw_exp2(exp_fe * hw_log2(fmaxf(Tbf - tm, NZ)));
      const float freeze = fminf(pot_freeze, wis);
      wis -= freeze;
      sog += freeze;
      const float melt = fminf(fmaxf(ddf * (tm - Tbm), 0.f), sog + snow);
      sog = sog + snow - melt;
      const float retention = wrf * sog;
      const float wtmp  = wis + melt + rain;
      const float avail = fmaxf(wtmp - retention, 0.f);
      wis = (avail > 0.f) ? retention : wtmp;

      // soil buckets
      const float ret_pot = ETe * pet;
      const float ratio   = vad * inv_vmax;
      float       ht0     = c_run * ratio * avail;
      const float infil   = avail - ht0;
      const float ht1     = c_v2p * ratio * vad;
      const float ht2     = c_vad * vad;
      const float ht3     = c_phr * phr;
      const float aet     = fminf(ret_pot, vad);
      const float vnew    = vad + infil - aet - ht1 - ht2;
      const float overflow = fmaxf(vnew - vmax, 0.f);
      vad = fminf(fmaxf(vnew, NZ), vmax);
      ht0 += overflow;
      phr = fmaxf(phr + ht1 - ht3, NZ);

      // fused 15-tap routing conv: one packed FMA per tap
      f32x2 qv; qv.x = ht0; qv.y = ht2 + ht3;
      const f32x2 d0 = pk_fma(wp[0], qv, acc[j]);    // tap l=0 folded in
      const float val = d0.x + d0.y;
      acc[j] = (f32x2)0.f;
#pragma unroll
      for (int l = 1; l < LENF; ++l)
        acc[(j + l) % LENF] = pk_fma(wp[l], qv, acc[(j + l) % LENF]);

      if (valid) out[(size_t)(tbase + j) * G_CELLS + g] = val;
    }
    __syncthreads();   // all waves done reading tile[k&1] before reuse
  }
}

extern "C" void kernel_launch(void* const* d_in, const int* in_sizes, int n_in,
                              void* d_out, int out_size, void* d_ws,
                              size_t ws_size, hipStream_t stream) {
  (void)in_sizes; (void)n_in; (void)out_size; (void)d_ws; (void)ws_size;
  const float* x_phy  = (const float*)d_in[0];
  const float* params = (const float*)d_in[1];
  float*       out    = (float*)d_out;
  const int grid = (G_CELLS + BLOCK - 1) / BLOCK;   // 118 blocks x 4 waves
  hipLaunchKernelGGL(exphydro_kernel, dim3(grid), dim3(BLOCK), 0, stream,
                     x_phy, params, out);
}